// GCN_Baseline_85418309583489
// MI455X (gfx1250) — compile-verified
//
#include <hip/hip_runtime.h>
#include <hip/hip_bf16.h>

typedef __attribute__((ext_vector_type(2))) float v2f;
typedef __attribute__((ext_vector_type(8))) float v8f;

#define N_NODES 50000
#define N_EDGES 800000
#define F_IN    96
#define F_H     96
#define F_OUT   40

// ---------------- utility kernels ----------------

__global__ void zero_f32(float* __restrict__ p, long n) {
    long i = (long)blockIdx.x * blockDim.x + threadIdx.x;
    long stride = (long)gridDim.x * blockDim.x;
    for (; i < n; i += stride) p[i] = 0.0f;
}

__global__ void copy_f32x4(const float4* __restrict__ in, float4* __restrict__ out, long n4) {
    long i = (long)blockIdx.x * blockDim.x + threadIdx.x;
    long stride = (long)gridDim.x * blockDim.x;
    for (; i < n4; i += stride) out[i] = in[i];
}

// deg[v] += 1 for src and dst index streams
__global__ void degree_kernel(const int* __restrict__ src, const int* __restrict__ dst,
                              float* __restrict__ deg_src, float* __restrict__ deg_dst) {
    int e = blockIdx.x * blockDim.x + threadIdx.x;
    if (e < N_EDGES) {
        atomicAdd(&deg_src[src[e]], 1.0f);
        atomicAdd(&deg_dst[dst[e]], 1.0f);
    }
}

// deg -> rsqrt(max(deg,1))  (DGL norm='both', zero-degree clamped)
__global__ void norm_kernel(float* __restrict__ deg, int n) {
    int i = blockIdx.x * blockDim.x + threadIdx.x;
    if (i < n) deg[i] = rsqrtf(fmaxf(deg[i], 1.0f));
}

// agg[dst, :] += h[src, :] * nsrc[src]
// One wave32 per edge; F = 96 = 3 features per lane. Edge indices and the
// source-side norm are loaded once per wave; gathers are coalesced 128B bursts;
// scatter uses non-returning global_atomic_add_f32.
__global__ void scatter_edge_kernel(const float* __restrict__ h,
                                    const int* __restrict__ src, const int* __restrict__ dst,
                                    const float* __restrict__ nsrc,
                                    float* __restrict__ agg) {
    const int e = blockIdx.x * (blockDim.x >> 5) + (threadIdx.x >> 5);
    if (e >= N_EDGES) return;                 // wave-uniform exit
    const int lane = threadIdx.x & 31;
    const int s = src[e];
    const int d = dst[e];
    const float ns = nsrc[s];
    const float* __restrict__ hs = h   + (long)s * F_IN;
    float*       __restrict__ ad = agg + (long)d * F_IN;
    #pragma unroll
    for (int j = 0; j < 3; ++j) {
        const int f = lane + 32 * j;
        atomicAdd(&ad[f], hs[f] * ns);
    }
}

// ---------------- WMMA GEMM: out = relu?((A .* ndst_row) @ W + b) ----------------
// A: M x K (row-major, M % 16 == 0, K % 4 == 0), W: K x N, out: M x N
// One wave32 computes a 16x16 output tile via V_WMMA_F32_16X16X4_F32 chained over K.
__global__ void gemm_wmma_f32(const float* __restrict__ A,
                              const float* __restrict__ ndst,
                              const float* __restrict__ W,
                              const float* __restrict__ bias,
                              float* __restrict__ out,
                              int K, int N, int tiles_n, int total_tiles, int do_relu) {
    __shared__ float sW[F_IN * F_H];   // 96*96*4 = 36 KB, worst case

    // Cooperative stage of W into LDS (all threads, before any early-out).
    for (int i = threadIdx.x; i < K * N; i += blockDim.x) sW[i] = W[i];
    __syncthreads();

    const int gw = blockIdx.x * (blockDim.x >> 5) + (threadIdx.x >> 5);
    if (gw >= total_tiles) return;          // wave-uniform exit; EXEC stays all-1s below

    const int mt = gw / tiles_n;
    const int nt = gw - mt * tiles_n;

    const int lane = threadIdx.x & 31;
    const int half = lane >> 4;             // 0: K pair {0,1}; 1: K pair {2,3}
    const int l16  = lane & 15;

    const int row  = mt * 16 + l16;         // A-operand row for this lane
    const float nd = ndst[row];
    const float* __restrict__ arow = A + (long)row * K;

    const int colg = nt * 16 + l16;         // B/D column for this lane
    const int colc = (colg < N) ? colg : (N - 1);   // clamp (only matters when N=40)

    v8f c;
    const float bv = bias[colc];
    #pragma unroll
    for (int r = 0; r < 8; ++r) c[r] = bv;  // D = A@B + C with C preloaded with bias

    for (int k0 = 0; k0 < K; k0 += 4) {
        const int ka = k0 + 2 * half;
        v2f a, b;
        a.x = arow[ka]     * nd;            // fold norm_dst into A rows
        a.y = arow[ka + 1] * nd;
        b.x = sW[ka * N + colc];
        b.y = sW[(ka + 1) * N + colc];
        c = __builtin_amdgcn_wmma_f32_16x16x4_f32(
                false, a, false, b, (short)0, c, false, false);
    }

    if (colg < N) {
        #pragma unroll
        for (int r = 0; r < 8; ++r) {
            float v = c[r];
            if (do_relu) v = fmaxf(v, 0.0f);
            // C/D layout: VGPR r -> M = r (lanes 0-15) or r+8 (lanes 16-31)
            out[(long)(mt * 16 + r + 8 * half) * N + colg] = v;
        }
    }
}

// ---------------- host launcher ----------------

extern "C" void kernel_launch(void* const* d_in, const int* in_sizes, int n_in,
                              void* d_out, int out_size, void* d_ws, size_t ws_size,
                              hipStream_t stream) {
    const float* x  = (const float*)d_in[0];
    const float* W1 = (const float*)d_in[1];
    const float* b1 = (const float*)d_in[2];
    const float* W2 = (const float*)d_in[3];
    const float* b2 = (const float*)d_in[4];
    const int*  src = (const int*)d_in[5];
    const int*  dst = (const int*)d_in[6];

    float* out = (float*)d_out;
    float* h2  = out;                                   // 50000 x 40
    float* h1  = out + (long)N_NODES * F_OUT;           // 50000 x 96
    float* xo  = h1  + (long)N_NODES * F_H;             // 50000 x 96 (pass-through x)

    float* ws   = (float*)d_ws;
    float* nsrc = ws;                                   // 50000
    float* ndst = ws + N_NODES;                         // 50000
    float* agg  = ws + 2 * N_NODES;                     // 50000 x 96 (reused both layers)

    const int tiles_m = N_NODES / 16;                   // 3125 (exact)
    const int edge_blocks = (N_EDGES + 7) / 8;          // 8 waves (edges) per 256-thread block

    // 0) zero norms + agg (ws is poisoned each run)
    zero_f32<<<2048, 256, 0, stream>>>(ws, (long)2 * N_NODES + (long)N_NODES * F_H);

    // pass-through output: copy x (independent of everything else)
    copy_f32x4<<<2048, 256, 0, stream>>>((const float4*)x, (float4*)xo,
                                         (long)N_NODES * F_IN / 4);

    // 1) degrees -> norms
    degree_kernel<<<(N_EDGES + 255) / 256, 256, 0, stream>>>(src, dst, nsrc, ndst);
    norm_kernel<<<(2 * N_NODES + 255) / 256, 256, 0, stream>>>(ws, 2 * N_NODES);

    // 2) layer 1 aggregation: agg[dst] += x[src] * nsrc[src]
    scatter_edge_kernel<<<edge_blocks, 256, 0, stream>>>(x, src, dst, nsrc, agg);

    // 3) h1 = relu((agg .* ndst) @ W1 + b1)
    {
        int tn = F_H / 16;                              // 6
        int tot = tiles_m * tn;                         // 18750 tiles
        gemm_wmma_f32<<<(tot + 7) / 8, 256, 0, stream>>>(agg, ndst, W1, b1, h1,
                                                         F_IN, F_H, tn, tot, 1);
    }

    // 4) re-zero agg, layer 2 aggregation: agg[dst] += h1[src] * nsrc[src]
    zero_f32<<<2048, 256, 0, stream>>>(agg, (long)N_NODES * F_H);
    scatter_edge_kernel<<<edge_blocks, 256, 0, stream>>>(h1, src, dst, nsrc, agg);

    // 5) h2 = (agg .* ndst) @ W2 + b2
    {
        int tn = (F_OUT + 15) / 16;                     // 3 (cols 40..47 masked)
        int tot = tiles_m * tn;                         // 9375 tiles
        gemm_wmma_f32<<<(tot + 7) / 8, 256, 0, stream>>>(agg, ndst, W2, b2, h2,
                                                         F_H, F_OUT, tn, tot, 0);
    }
}